// ShifthWiseConv1dImplicit_37830071943695
// MI455X (gfx1250) — compile-verified
//
#include <hip/hip_runtime.h>

// ShiftWise depthwise conv + 3x BatchNorm + residual + ghost passthrough for
// MI455X (gfx1250, wave32). Memory-bound (~375 MB traffic -> ~16us @ 23.3TB/s).
// Conv runs on the matrix pipe via V_WMMA_F32_16X16X4_F32 (f32 precision):
//   D(16x16) = A(16x4) * B(4x16) + C chained over K=16 (15 taps + zero pad)
//   A rows 0..2 = x1/x2/x3 effective filters, rows 3..15 = 0
//   B columns   = input windows of 16 consecutive output positions (LDS)
// Lane n<16 ends with x1,x2,x3 of position n in C VGPRs 0..2; lanes 16..31
// hold zero rows (8..10). Boundary-band corrections are hoisted out of the
// hot loop. Row staging uses GLOBAL_LOAD_ASYNC_TO_LDS_B128 (ASYNCcnt path).

#define TPB 256
#define WPB 8          // waves per block (wave32)
constexpr int BATCH = 32;
constexpr int INCH  = 256;
constexpr int LEN   = 4096;
constexpr int REPN  = 198;   // 256 - int(256*0.23)
constexpr int GHN   = 58;
constexpr int HALO  = 8;
constexpr int TILES = LEN / 16;
constexpr int ITERS = TILES / WPB;   // 32, compile-time trip count
constexpr float INV_CNT = 1.0f / (float)(BATCH * LEN);
constexpr float EPSV = 1e-5f;

typedef float v2f __attribute__((ext_vector_type(2)));
typedef float v8f __attribute__((ext_vector_type(8)));
typedef int v4i_vs __attribute__((vector_size(16)));   // matches builtin param

#define AS1 __attribute__((address_space(1)))
#define AS3 __attribute__((address_space(3)))

// ---------------------------------------------------------------------------
// MT19937 matching legacy numpy (np.random.seed(int) -> mt19937_seed)
// ---------------------------------------------------------------------------
__device__ inline unsigned mt_next(unsigned* mt, int& mti) {
  if (mti >= 624) {
    for (int i = 0; i < 624; ++i) {
      unsigned y = (mt[i] & 0x80000000u) | (mt[(i + 1) % 624] & 0x7fffffffu);
      mt[i] = mt[(i + 397) % 624] ^ (y >> 1) ^ ((y & 1u) ? 0x9908b0dfu : 0u);
    }
    mti = 0;
  }
  unsigned y = mt[mti++];
  y ^= y >> 11;
  y ^= (y << 7) & 0x9d2c5680u;
  y ^= (y << 15) & 0xefc60000u;
  y ^= y >> 18;
  return y;
}

// np.random.seed(123); ghost = sort(choice(256,58,replace=False)); rep = rest.
__global__ void setup_indices(int* __restrict__ rep, int* __restrict__ ghost) {
  __shared__ unsigned mt[624];
  __shared__ short perm[INCH];
  __shared__ unsigned char isg[INCH];
  if (threadIdx.x != 0) return;
  mt[0] = 123u;
  for (int i = 1; i < 624; ++i)
    mt[i] = 1812433253u * (mt[i - 1] ^ (mt[i - 1] >> 30)) + (unsigned)i;
  int mti = 624;
  for (int i = 0; i < INCH; ++i) { perm[i] = (short)i; isg[i] = 0; }
  for (int i = INCH - 1; i > 0; --i) {
    unsigned msk = (unsigned)i;
    msk |= msk >> 1; msk |= msk >> 2; msk |= msk >> 4;
    msk |= msk >> 8; msk |= msk >> 16;
    unsigned j;
    do { j = mt_next(mt, mti) & msk; } while (j > (unsigned)i);
    short tmp = perm[i]; perm[i] = perm[j]; perm[j] = tmp;
  }
  for (int k = 0; k < GHN; ++k) isg[perm[k]] = 1;
  int gi = 0, ri = 0;
  for (int v = 0; v < INCH; ++v) {
    if (isg[v]) ghost[gi++] = v; else rep[ri++] = v;
  }
}

// ---------------------------------------------------------------------------
// Conv helpers.  wr[15] = w[c,i,k] flattened (i*3+k).  Tap j sits at l+j-7.
// W1[j]=wr[j]; W2[j]=wr[(4-j/3)*3+j%3]; W3[j]=wr[j] only for j in [6,9).
// ---------------------------------------------------------------------------
__device__ __forceinline__ float wa(const float* wr, int m, int j) {
  if (j >= 15 || m >= 3) return 0.f;
  if (m == 0) return wr[j];
  if (m == 1) return wr[(4 - j / 3) * 3 + (j % 3)];
  return (j >= 6 && j < 9) ? wr[j] : 0.f;
}

__device__ __forceinline__ void build_a(const float* wr, int lane, v2f a[4]) {
  int m = lane & 15, half = lane >> 4;
#pragma unroll
  for (int t = 0; t < 4; ++t) {
    int j0 = 4 * t + 2 * half;
    a[t][0] = wa(wr, m, j0);
    a[t][1] = wa(wr, m, j0 + 1);
  }
}

__device__ __forceinline__ v8f conv_tile(const float* srow, const v2f a[4],
                                         int pos0, int n, int half) {
  const float* xp = srow + (HALO - 7) + pos0 + n;
  v8f d = {};
#pragma unroll
  for (int t = 0; t < 4; ++t) {
    int kb = 4 * t + 2 * half;
    v2f bv;
    bv[0] = xp[kb];
    bv[1] = xp[kb + 1];
    d = __builtin_amdgcn_wmma_f32_16x16x4_f32(false, a[t], false, bv,
                                              (short)0, d, false, false);
  }
  return d;
}

// Boundary corrections: _shift() zeroes whole 3-tap bands whose band-center
// leaves [0,L); the pure 15-tap conv keeps one spurious tap at 4 positions.
__constant__ int c_pos[4] = {2, 5, LEN - 6, LEN - 3};
__constant__ int c_i1[4]  = {5, 2, 12, 9};    // wr index of x1 correction
__constant__ int c_i2[4]  = {11, 14, 0, 3};   // wr index of x2 correction

// ---------------------------------------------------------------------------
// Row staging: async global->LDS (ASYNCcnt path) with vector-copy fallback
// ---------------------------------------------------------------------------
__device__ __forceinline__ void load_row(float* srow, const float* src) {
  int t = threadIdx.x;
  if (t < HALO) { srow[t] = 0.f; srow[HALO + LEN + t] = 0.f; }
#if __has_builtin(__builtin_amdgcn_global_load_async_to_lds_b128)
#pragma unroll
  for (int i = 0; i < (LEN / 4) / TPB; ++i) {
    int idx = 4 * (t + i * TPB);
    __builtin_amdgcn_global_load_async_to_lds_b128(
        (AS1 v4i_vs*)(src + idx), (AS3 v4i_vs*)(srow + HALO + idx), 0, 0);
  }
#if __has_builtin(__builtin_amdgcn_s_wait_asynccnt)
  __builtin_amdgcn_s_wait_asynccnt(0);
#else
  asm volatile("s_wait_asynccnt 0" ::: "memory");
#endif
#else
  const float4* s4 = (const float4*)src;
  float4* d4 = (float4*)(srow + HALO);
  for (int i = t; i < LEN / 4; i += TPB) d4[i] = s4[i];
#endif
}

__device__ __forceinline__ float wred(float v) {
#pragma unroll
  for (int off = 16; off > 0; off >>= 1) v += __shfl_down(v, off, 32);
  return v;
}

// ---------------------------------------------------------------------------
// Pass 1: per-(b,c) row -> partial {sum,sumsq} of x1,x2,x3 (deterministic)
// ---------------------------------------------------------------------------
__global__ __launch_bounds__(TPB) void shiftwise_stats(
    const float* __restrict__ in, const float* __restrict__ w1,
    const float* __restrict__ w2, const int* __restrict__ rep,
    float* __restrict__ partial) {
  __shared__ __align__(16) float srow[LEN + 2 * HALO];
  __shared__ float wr[16];
  __shared__ float red[WPB][6];
  int bx = blockIdx.x;
  int c = bx >> 5, b = bx & 31;
  int t = threadIdx.x;
  if (t < 15) wr[t] = w1[c * 15 + t] + w2[c * 15 + t];
  if (t == 15) wr[15] = 0.f;
  load_row(srow, in + ((size_t)b * INCH + rep[c]) * LEN);
  __syncthreads();

  int wave = t >> 5, lane = t & 31;
  int n = lane & 15, half = lane >> 4;
  v2f a[4];
  build_a(wr, lane, a);
  float s1 = 0, q1 = 0, s2 = 0, q2 = 0, s3 = 0, q3 = 0;
#pragma unroll 2
  for (int it = 0; it < ITERS; ++it) {   // uniform trip count: scalar loop
    int pos0 = (it * WPB + wave) << 4;
    v8f d = conv_tile(srow, a, pos0, n, half);
    // lanes 16..31 hold zero rows: unconditional accumulation is exact
    s1 += d[0]; q1 += d[0] * d[0];
    s2 += d[1]; q2 += d[1] * d[1];
    s3 += d[2]; q3 += d[2] * d[2];
  }
  s1 = wred(s1); q1 = wred(q1); s2 = wred(s2);
  q2 = wred(q2); s3 = wred(s3); q3 = wred(q3);
  if (lane == 0) {
    red[wave][0] = s1; red[wave][1] = q1; red[wave][2] = s2;
    red[wave][3] = q2; red[wave][4] = s3; red[wave][5] = q3;
  }
  __syncthreads();
  if (t == 0) {
    float acc[6];
#pragma unroll
    for (int s = 0; s < 6; ++s) {
      float a0 = 0.f;
      for (int w = 0; w < WPB; ++w) a0 += red[w][s];
      acc[s] = a0;
    }
    // hoisted boundary corrections (4 positions per row)
    float x0 = srow[HALO], xl = srow[HALO + LEN - 1];
#pragma unroll
    for (int e = 0; e < 4; ++e) {
      int p = c_pos[e];
      float x1p = 0.f, x2p = 0.f;
#pragma unroll
      for (int j = 0; j < 15; ++j) {
        float xv = srow[HALO + p + j - 7];
        x1p += wr[j] * xv;
        x2p += wr[(4 - j / 3) * 3 + (j % 3)] * xv;
      }
      float src = (e < 2) ? x0 : xl;
      float c1 = src * wr[c_i1[e]], c2 = src * wr[c_i2[e]];
      acc[0] -= c1; acc[1] += c1 * c1 - 2.f * x1p * c1;
      acc[2] -= c2; acc[3] += c2 * c2 - 2.f * x2p * c2;
    }
#pragma unroll
    for (int s = 0; s < 6; ++s) partial[(size_t)bx * 6 + s] = acc[s];
  }
}

// Fixed-order per-channel reduction over the 32 batch rows.
__global__ void shiftwise_reduce(const float* __restrict__ partial,
                                 float* __restrict__ stats) {
  int gid = blockIdx.x * blockDim.x + threadIdx.x;
  if (gid >= REPN * 6) return;
  int c = gid / 6, s = gid % 6;
  float acc = 0.f;
  for (int b = 0; b < BATCH; ++b)
    acc += partial[((size_t)(c * BATCH + b)) * 6 + s];
  stats[gid] = acc;
}

// ---------------------------------------------------------------------------
// Pass 2: recompute convs, fused 3x BN + residual, staged b128 stores
// ---------------------------------------------------------------------------
__global__ __launch_bounds__(TPB) void shiftwise_final(
    const float* __restrict__ in, const float* __restrict__ w1,
    const float* __restrict__ w2, const int* __restrict__ rep,
    const float* __restrict__ stats,
    const float* __restrict__ g1, const float* __restrict__ be1,
    const float* __restrict__ g2, const float* __restrict__ be2,
    const float* __restrict__ g3, const float* __restrict__ be3,
    float* __restrict__ out) {
  __shared__ __align__(16) float srow[LEN + 2 * HALO];
  __shared__ __align__(16) float orow[LEN + 16];  // +16 dump slots for half==1
  __shared__ float wr[16];
  int bx = blockIdx.x;
  int c = bx >> 5, b = bx & 31;
  int t = threadIdx.x;
  if (t < 15) wr[t] = w1[c * 15 + t] + w2[c * 15 + t];
  if (t == 15) wr[15] = 0.f;
  load_row(srow, in + ((size_t)b * INCH + rep[c]) * LEN);
  __syncthreads();

  float m1 = stats[c * 6 + 0] * INV_CNT;
  float va = stats[c * 6 + 1] * INV_CNT - m1 * m1;
  float m2 = stats[c * 6 + 2] * INV_CNT;
  float vb = stats[c * 6 + 3] * INV_CNT - m2 * m2;
  float m3 = stats[c * 6 + 4] * INV_CNT;
  float vc = stats[c * 6 + 5] * INV_CNT - m3 * m3;
  float sc1 = rsqrtf(va + EPSV) * g1[c];
  float sc2 = rsqrtf(vb + EPSV) * g2[c];
  float sc3 = rsqrtf(vc + EPSV) * g3[c];
  float ofs = be1[c] + be2[c] + be3[c] - m1 * sc1 - m2 * sc2 - m3 * sc3;

  int wave = t >> 5, lane = t & 31;
  int n = lane & 15, half = lane >> 4;
  v2f a[4];
  build_a(wr, lane, a);
#pragma unroll 2
  for (int it = 0; it < ITERS; ++it) {   // uniform trip count: scalar loop
    int pos = ((it * WPB + wave) << 4) + n;
    v8f d = conv_tile(srow, a, (it * WPB + wave) << 4, n, half);
    float val = d[0] * sc1 + d[1] * sc2 + d[2] * sc3 + ofs + srow[HALO + pos];
    int oaddr = half ? (LEN + n) : pos;  // branchless: cndmask to dump slots
    orow[oaddr] = val;
  }
  __syncthreads();
  if (t < 4) {  // hoisted boundary corrections
    float s = (t < 2) ? srow[HALO] : srow[HALO + LEN - 1];
    orow[c_pos[t]] -= s * wr[c_i1[t]] * sc1 + s * wr[c_i2[t]] * sc2;
  }
  __syncthreads();
  float4* dst4 = (float4*)(out + ((size_t)b * INCH + c) * LEN);
  const float4* s4 = (const float4*)orow;
  for (int i = t; i < LEN / 4; i += TPB) dst4[i] = s4[i];
}

__global__ __launch_bounds__(TPB) void ghost_copy(const float* __restrict__ in,
                                                  const int* __restrict__ ghost,
                                                  float* __restrict__ out) {
  int j = blockIdx.x % GHN, b = blockIdx.x / GHN;
  const float4* s = (const float4*)(in + ((size_t)b * INCH + ghost[j]) * LEN);
  float4* d = (float4*)(out + ((size_t)b * INCH + REPN + j) * LEN);
  for (int i = threadIdx.x; i < LEN / 4; i += TPB) d[i] = s[i];
}

// ---------------------------------------------------------------------------
extern "C" void kernel_launch(void* const* d_in, const int* in_sizes, int n_in,
                              void* d_out, int out_size, void* d_ws,
                              size_t ws_size, hipStream_t stream) {
  const float* in  = (const float*)d_in[0];
  const float* w1  = (const float*)d_in[1];
  const float* w2  = (const float*)d_in[2];
  const float* g1  = (const float*)d_in[3];
  const float* be1 = (const float*)d_in[4];
  const float* g2  = (const float*)d_in[5];
  const float* be2 = (const float*)d_in[6];
  const float* g3  = (const float*)d_in[7];
  const float* be3 = (const float*)d_in[8];
  float* out = (float*)d_out;

  // ws: rep[198] | ghost[58] (256 ints) | partial[6336*6] f32 | stats[198*6]
  int* rep   = (int*)d_ws;
  int* ghost = rep + REPN;
  float* partial = (float*)(rep + INCH);
  float* stats   = partial + (size_t)REPN * BATCH * 6;

  setup_indices<<<1, 32, 0, stream>>>(rep, ghost);
  shiftwise_stats<<<REPN * BATCH, TPB, 0, stream>>>(in, w1, w2, rep, partial);
  shiftwise_reduce<<<(REPN * 6 + 255) / 256, 256, 0, stream>>>(partial, stats);
  shiftwise_final<<<REPN * BATCH, TPB, 0, stream>>>(in, w1, w2, rep, stats,
                                                    g1, be1, g2, be2, g3, be3,
                                                    out);
  ghost_copy<<<BATCH * GHN, TPB, 0, stream>>>(in, ghost, out);
}